// GraphSAGE_38104949850571
// MI455X (gfx1250) — compile-verified
//
#include <hip/hip_runtime.h>

typedef __attribute__((ext_vector_type(2))) float v2f;
typedef __attribute__((ext_vector_type(8))) float v8f;

#define NNODES 50000
#define NEDGES 800000
#define FIN    64
#define HID    128
#define NGRAPH 512

// ---------------- zero fill (grid-stride) ----------------
__global__ void k_zero(float* __restrict__ p, long n) {
  long i = (long)blockIdx.x * blockDim.x + threadIdx.x;
  long stride = (long)gridDim.x * blockDim.x;
  for (; i < n; i += stride) p[i] = 0.0f;
}

// ---------------- combine weights/biases ----------------
__global__ void k_prep(const float* __restrict__ wr0, const float* __restrict__ r0w,
                       const float* __restrict__ bl0, const float* __restrict__ r0b,
                       const float* __restrict__ wr1, const float* __restrict__ r1w,
                       const float* __restrict__ bl1, const float* __restrict__ r1b,
                       float* __restrict__ wc0, float* __restrict__ b0,
                       float* __restrict__ wc1, float* __restrict__ b1) {
  int i = blockIdx.x * 256 + threadIdx.x;
  if (i < FIN * HID)  wc0[i] = wr0[i] + r0w[i];
  if (i < HID * HID)  wc1[i] = wr1[i] + r1w[i];
  if (i < HID) { b0[i] = bl0[i] + r0b[i]; b1[i] = bl1[i] + r1b[i]; }
}

// ---------------- pack [Wl ; Wc] (KT x 128) into WMMA-fragment order ----------------
// Fragment layout: flat[ ((wv*(KT/4) + q)*32 + lane)*2 + {0,1} ] =
//   B[4q + 2*(lane>>4) + {0,1}][wv*16 + (lane&15)]
// -> per k-step each wave does ONE coalesced global_load_b64 (256B/wave).
template <int KT>
__global__ void k_pack(const float* __restrict__ Wl, const float* __restrict__ Wc,
                       float* __restrict__ dst) {
  constexpr int Q = KT / 4;
  int idx = blockIdx.x * 256 + threadIdx.x;
  if (idx >= KT * HID) return;
  int which   = idx & 1;
  int pairIdx = idx >> 1;
  int lane    = pairIdx & 31;
  int t       = pairIdx >> 5;
  int q       = t % Q;
  int wv      = t / Q;
  int hh      = lane >> 4;
  int n       = lane & 15;
  int k       = 4 * q + 2 * hh + which;      // row in stacked [Wl;Wc]
  int c       = wv * 16 + n;                 // column
  float v = (k < KT / 2) ? Wl[k * HID + c] : Wc[(k - KT / 2) * HID + c];
  dst[idx] = v;
}

// ---------------- in-degree ----------------
__global__ void k_deg(const int* __restrict__ dst, float* __restrict__ deg) {
  int e = blockIdx.x * 256 + threadIdx.x;
  if (e < NEDGES) atomicAdd(&deg[dst[e]], 1.0f);
}

// ---------------- edge scatter-add (push, L2-resident targets) ----------------
template <int LOG2C>
__global__ void k_scatter(const float* __restrict__ feat, const int* __restrict__ src,
                          const int* __restrict__ dst, float* __restrict__ agg) {
  constexpr int F = (1 << LOG2C) * 4;
  long idx = (long)blockIdx.x * 256 + threadIdx.x;
  int e = (int)(idx >> LOG2C);
  int c = (int)(idx & ((1 << LOG2C) - 1));
  if (e >= NEDGES) return;
  int s = src[e], d = dst[e];
  float4 v = *(const float4*)(feat + (size_t)s * F + c * 4);
  float* out = agg + (size_t)d * F + c * 4;
  atomicAdd(out + 0, v.x);
  atomicAdd(out + 1, v.y);
  atomicAdd(out + 2, v.z);
  atomicAdd(out + 3, v.w);
}

// ---------------- fused dual-GEMM + bias + LayerNorm (+ReLU) ----------------
// A = [agg*inv_deg | self] (16 x 2K) staged in padded LDS; B pre-packed in
// fragment order; single K-loop of V_WMMA_F32_16X16X4_F32.
template <int K, bool RELU>
__global__ __launch_bounds__(256)
void k_gemm_ln(const float* __restrict__ agg, const float* __restrict__ self,
               const float* __restrict__ Bf, const float* __restrict__ bias,
               const float* __restrict__ lng, const float* __restrict__ lnb,
               const float* __restrict__ deg, float* __restrict__ out) {
  constexpr int KT = 2 * K;
  constexpr int Q  = KT / 4;
  __shared__ float At[16][KT + 2];   // +2 pad: avoid 16-way LDS bank conflict
  __shared__ float Ot[16][HID + 1];
  __shared__ float sInv[16];

  const int tid = threadIdx.x;
  const int i0  = blockIdx.x * 16;

  if (tid < 16) sInv[tid] = 1.0f / fmaxf(deg[i0 + tid], 1.0f);
  __syncthreads();

  for (int idx = tid; idx < 16 * KT; idx += 256) {
    int row = idx / KT, col = idx - row * KT;
    float v;
    if (col < K) v = agg[(size_t)(i0 + row) * K + col] * sInv[row];
    else         v = self[(size_t)(i0 + row) * K + (col - K)];
    At[row][col] = v;
  }
  __syncthreads();

  const int lane = tid & 31;
  const int wv   = tid >> 5;
  const int n    = lane & 15;            // A row / D col
  const int kk   = (lane >> 4) * 2;      // K sub-offset per lane half
  const int col0 = wv * 16;

  const float2* bp = (const float2*)Bf + (size_t)(wv * Q) * 32 + lane;

  v8f acc = {0.f, 0.f, 0.f, 0.f, 0.f, 0.f, 0.f, 0.f};
#pragma unroll 4
  for (int q = 0; q < Q; ++q) {
    float2 bb = bp[q * 32];              // one coalesced b64 per WMMA
    v2f a, b;
    a.x = At[n][4 * q + kk];
    a.y = At[n][4 * q + kk + 1];
    b.x = bb.x;
    b.y = bb.y;
    acc = __builtin_amdgcn_wmma_f32_16x16x4_f32(false, a, false, b,
                                                (short)0, acc, false, false);
  }

  // spill D tile (+bias) to LDS for cross-wave LayerNorm
  const int mh = (lane >> 4) * 8;
  const float bcol = bias[col0 + n];
#pragma unroll
  for (int r = 0; r < 8; ++r) Ot[mh + r][col0 + n] = acc[r] + bcol;
  __syncthreads();

  // LayerNorm: 16 threads per node, butterfly reduce within 16-lane groups
  const int node = tid >> 4, j = tid & 15;
  float s = 0.f, sq = 0.f;
#pragma unroll
  for (int i = 0; i < 8; ++i) {
    float v = Ot[node][j * 8 + i];
    s += v; sq += v * v;
  }
  for (int off = 8; off > 0; off >>= 1) {
    s  += __shfl_xor(s, off, 16);
    sq += __shfl_xor(sq, off, 16);
  }
  const float mu   = s * (1.0f / HID);
  const float var  = sq * (1.0f / HID) - mu * mu;
  const float rstd = rsqrtf(var + 1e-5f);
#pragma unroll
  for (int i = 0; i < 8; ++i) {
    int c = j * 8 + i;
    float v = (Ot[node][c] - mu) * rstd * lng[c] + lnb[c];
    if (RELU) v = fmaxf(v, 0.0f);
    out[(size_t)(i0 + node) * HID + c] = v;
  }
}

// ---------------- pooled readout: per-node dot, then segment sum ----------------
__global__ void k_pool(const float* __restrict__ h1, const int* __restrict__ batch,
                       const float* __restrict__ ow, float* __restrict__ gsum,
                       float* __restrict__ cnt) {
  long idx = (long)blockIdx.x * 256 + threadIdx.x;   // N*16 threads exactly
  int node = (int)(idx >> 4);
  int j = (int)(idx & 15);
  const float* row = h1 + (size_t)node * HID + j * 8;
  const float* w   = ow + j * 8;
  float s = 0.f;
#pragma unroll
  for (int i = 0; i < 8; ++i) s += row[i] * w[i];
  for (int off = 8; off > 0; off >>= 1) s += __shfl_xor(s, off, 16);
  if (j == 0) {
    int b = batch[node];
    atomicAdd(&gsum[b], s);
    atomicAdd(&cnt[b], 1.0f);
  }
}

__global__ void k_final(const float* __restrict__ gsum, const float* __restrict__ cnt,
                        const float* __restrict__ ob, float* __restrict__ out) {
  int g = blockIdx.x * 256 + threadIdx.x;
  if (g < NGRAPH) out[g] = gsum[g] / fmaxf(cnt[g], 1.0f) + ob[0];
}

// ---------------- launcher ----------------
extern "C" void kernel_launch(void* const* d_in, const int* in_sizes, int n_in,
                              void* d_out, int out_size, void* d_ws, size_t ws_size,
                              hipStream_t stream) {
  const float* x        = (const float*)d_in[0];
  const int*   ei       = (const int*)d_in[1];
  const int*   batch    = (const int*)d_in[2];
  const float* lin_l0_w = (const float*)d_in[3];
  const float* lin_l0_b = (const float*)d_in[4];
  const float* lin_r0_w = (const float*)d_in[5];
  const float* res0_w   = (const float*)d_in[6];
  const float* res0_b   = (const float*)d_in[7];
  const float* ln0_g    = (const float*)d_in[8];
  const float* ln0_b    = (const float*)d_in[9];
  const float* lin_l1_w = (const float*)d_in[10];
  const float* lin_l1_b = (const float*)d_in[11];
  const float* lin_r1_w = (const float*)d_in[12];
  const float* res1_w   = (const float*)d_in[13];
  const float* res1_b   = (const float*)d_in[14];
  const float* ln1_g    = (const float*)d_in[15];
  const float* ln1_b    = (const float*)d_in[16];
  const float* out_w    = (const float*)d_in[17];
  const float* out_b    = (const float*)d_in[18];

  const int* src = ei;
  const int* dst = ei + NEDGES;

  // workspace layout (floats); zero-init region first and contiguous
  float* ws = (float*)d_ws;
  size_t o = 0;
  float* deg  = ws + o; o += NNODES;
  float* agg0 = ws + o; o += (size_t)NNODES * FIN;
  float* agg1 = ws + o; o += (size_t)NNODES * HID;
  float* gsum = ws + o; o += NGRAPH;
  float* cnt  = ws + o; o += NGRAPH;
  const long zeroN = (long)o;
  float* wc0  = ws + o; o += FIN * HID;
  float* b0   = ws + o; o += HID;
  float* wc1  = ws + o; o += HID * HID;
  float* b1   = ws + o; o += HID;
  float* bf0  = ws + o; o += 2 * FIN * HID;        // packed [Wl0 ; Wc0]
  float* bf1  = ws + o; o += 2 * HID * HID;        // packed [Wl1 ; Wc1]
  float* h0   = ws + o; o += (size_t)NNODES * HID;
  float* h1   = ws + o; o += (size_t)NNODES * HID;

  float* out = (float*)d_out;

  // 1. zero accumulators
  k_zero<<<4096, 256, 0, stream>>>(ws, zeroN);
  // 2. combined weights / biases, then pack into WMMA fragment order
  k_prep<<<(HID * HID + 255) / 256, 256, 0, stream>>>(
      lin_r0_w, res0_w, lin_l0_b, res0_b,
      lin_r1_w, res1_w, lin_l1_b, res1_b, wc0, b0, wc1, b1);
  k_pack<2 * FIN><<<(2 * FIN * HID) / 256, 256, 0, stream>>>(lin_l0_w, wc0, bf0);
  k_pack<2 * HID><<<(2 * HID * HID) / 256, 256, 0, stream>>>(lin_l1_w, wc1, bf1);
  // 3. in-degree
  k_deg<<<(NEDGES + 255) / 256, 256, 0, stream>>>(dst, deg);
  // 4. layer-0 aggregation (E*16 threads)
  k_scatter<4><<<(NEDGES * 16) / 256, 256, 0, stream>>>(x, src, dst, agg0);
  // 5. layer-0 dual-GEMM + LN + ReLU  (3125 row tiles)
  k_gemm_ln<FIN, true><<<NNODES / 16, 256, 0, stream>>>(
      agg0, x, bf0, b0, ln0_g, ln0_b, deg, h0);
  // 6. layer-1 aggregation (E*32 threads)
  k_scatter<5><<<(NEDGES * 32) / 256, 256, 0, stream>>>(h0, src, dst, agg1);
  // 7. layer-1 dual-GEMM + LN
  k_gemm_ln<HID, false><<<NNODES / 16, 256, 0, stream>>>(
      agg1, h0, bf1, b1, ln1_g, ln1_b, deg, h1);
  // 8. per-node readout dot + segment sum
  k_pool<<<(NNODES * 16) / 256, 256, 0, stream>>>(h1, batch, out_w, gsum, cnt);
  // 9. segment mean + final bias
  k_final<<<(NGRAPH + 255) / 256, 256, 0, stream>>>(gsum, cnt, out_b, out);
}